// PGDLoss_41841571397806
// MI455X (gfx1250) — compile-verified
//
#include <hip/hip_runtime.h>

// ---------------------------------------------------------------------------
// PGD entropy-knapsack loss on MI455X (gfx1250, wave32).
//
// One wave handles 16 rows. v[16][1000] lives in LDS in a WMMA-A-fragment
// layout. For pair j (8 columns), lane L (row = L&15, h = L>>4) holds 4
// floats at V[j*128 + L*4] which are global columns [8j+4h .. 8j+4h+3]:
//   floats 0,1 -> tile 2j, floats 2,3 -> tile 2j+1 (K = 2h, 2h+1).
// Bisection sum s[row] = sum_col clip(v - mid, 0, 1) is D = A x ones via
// V_WMMA_F32_16X16X4_F32 (exact f32); clip folds to one v_sub_f32 ... clamp,
// so VALU does 1 op/element while the reduction adds ride the matrix pipe.
// Pair 125 is -1e30 padding (clips to 0, excluded from min/max) so the
// bisect loop is an even 126 pairs with a fully static acc0..acc3 rotation.
// Row-sum extraction is a register mux + shfl_xor(16): no LDS, no barriers.
// ---------------------------------------------------------------------------

typedef float v2f __attribute__((ext_vector_type(2)));
typedef float v8f __attribute__((ext_vector_type(8)));

#define NCLS 1000
#define PAIRS 125            // real pairs; pair 125 is padding
#define PAIRS_PAD 126
#define KF 5.0f
#define ALPHAF 0.025f
#define N_ITER 80
#define N_BISECT 50

__device__ __forceinline__ float clamp01(float x) {
  return __builtin_amdgcn_fmed3f(x, 0.0f, 1.0f);  // folds to v_sub ... clamp
}

__device__ __forceinline__ v8f wmma_acc(v2f a, v2f b, v8f c) {
  return __builtin_amdgcn_wmma_f32_16x16x4_f32(false, a, false, b, (short)0, c,
                                               false, false);
}

// Elementwise sweep: p = (FIRST ? k/n : clip(v_old - nu)), pc = clip(p,eps),
// v = p + alpha*(c + ln(1-pc) - ln(pc)); track per-row min/max -> lo/hi.
template <bool FIRST>
__device__ __forceinline__ void sweep(float* __restrict__ V,
                                      const float* __restrict__ xrow, int lane,
                                      int h, float rnorm, float nu, float& lo,
                                      float& hi) {
  float vmin = 3.0e38f, vmax = -3.0e38f;
  for (int j = 0; j < PAIRS; ++j) {
    const float4 xc = *(const float4*)(xrow + 8 * j + 4 * h);
    const float c4[4] = {xc.x * rnorm, xc.y * rnorm, xc.z * rnorm,
                         xc.w * rnorm};
    float* slot = &V[j * 128 + lane * 4];
    float p4[4];
    if (FIRST) {
      p4[0] = p4[1] = p4[2] = p4[3] = KF / (float)NCLS;  // p0 = k/n
    } else {
      const float4 vo = *(const float4*)slot;
      p4[0] = clamp01(vo.x - nu);
      p4[1] = clamp01(vo.y - nu);
      p4[2] = clamp01(vo.z - nu);
      p4[3] = clamp01(vo.w - nu);
    }
    float vn[4];
#pragma unroll
    for (int k = 0; k < 4; ++k) {
      const float p  = p4[k];
      const float pc = fminf(fmaxf(p, 1e-6f), 1.0f - 1e-6f);
      const float g  = c4[k] + __logf(1.0f - pc) - __logf(pc);
      const float v  = fmaf(ALPHAF, g, p);
      vmin = fminf(vmin, v);
      vmax = fmaxf(vmax, v);
      vn[k] = v;
    }
    *(float4*)slot = make_float4(vn[0], vn[1], vn[2], vn[3]);
  }
  vmin = fminf(vmin, __shfl_xor(vmin, 16, 32));
  vmax = fmaxf(vmax, __shfl_xor(vmax, 16, 32));
  lo = vmin - 1.0f;
  hi = vmax;
}

// 50-step bisection; returns nu = 0.5*(lo+hi). Barrier-free.
__device__ __forceinline__ float bisect50(const float* __restrict__ V,
                                          int lane, bool b0, bool b1, bool b2,
                                          bool local, float lo, float hi) {
  const v2f bones = {1.0f, 1.0f};  // B = ones(4x16)
  for (int b = 0; b < N_BISECT; ++b) {
    const float mid = 0.5f * (lo + hi);
    v8f acc0 = {0.f, 0.f, 0.f, 0.f, 0.f, 0.f, 0.f, 0.f};
    v8f acc1 = {0.f, 0.f, 0.f, 0.f, 0.f, 0.f, 0.f, 0.f};
    v8f acc2 = {0.f, 0.f, 0.f, 0.f, 0.f, 0.f, 0.f, 0.f};
    v8f acc3 = {0.f, 0.f, 0.f, 0.f, 0.f, 0.f, 0.f, 0.f};
#pragma unroll 7
    for (int jp = 0; jp < PAIRS_PAD; jp += 2) {  // static acc rotation
      const float4 va = *(const float4*)&V[jp * 128 + lane * 4];
      const float4 vb = *(const float4*)&V[(jp + 1) * 128 + lane * 4];
      v2f a0 = {clamp01(va.x - mid), clamp01(va.y - mid)};
      v2f a1 = {clamp01(va.z - mid), clamp01(va.w - mid)};
      v2f a2 = {clamp01(vb.x - mid), clamp01(vb.y - mid)};
      v2f a3 = {clamp01(vb.z - mid), clamp01(vb.w - mid)};
      acc0 = wmma_acc(a0, bones, acc0);
      acc1 = wmma_acc(a1, bones, acc1);
      acc2 = wmma_acc(a2, bones, acc2);
      acc3 = wmma_acc(a3, bones, acc3);
    }
    const v8f s8 = (acc0 + acc1) + (acc2 + acc3);
    // D layout: half-wave h holds rows 8h..8h+7 in regs 0..7 (all columns
    // identical since B = ones). Mux s8[row&7], swap halves, select.
    const float m01 = b0 ? s8[1] : s8[0];
    const float m23 = b0 ? s8[3] : s8[2];
    const float m45 = b0 ? s8[5] : s8[4];
    const float m67 = b0 ? s8[7] : s8[6];
    const float mA  = b1 ? m23 : m01;
    const float mB  = b1 ? m67 : m45;
    const float smine  = b2 ? mB : mA;
    const float sother = __shfl_xor(smine, 16, 32);
    const float s = local ? smine : sother;
    const bool big = s > KF;
    lo = big ? mid : lo;
    hi = big ? hi : mid;
  }
  return 0.5f * (lo + hi);
}

__global__ __launch_bounds__(32) void pgd_rows16(const float* __restrict__ x,
                                                 const int* __restrict__ y,
                                                 float* __restrict__ partial) {
  __shared__ __align__(16) float V[16 * 1008];  // 126 pairs * 128 floats

  const int lane   = threadIdx.x;  // 0..31
  const int row_in = lane & 15;
  const int h      = lane >> 4;
  const int grp    = blockIdx.x;
  const float* __restrict__ xrow = x + (size_t)(grp * 16 + row_in) * NCLS;

  // lane-constant mux controls for the row-sum extraction
  const bool b0 = (row_in & 1) != 0;
  const bool b1 = (row_in & 2) != 0;
  const bool b2 = (row_in & 4) != 0;
  const bool local = ((row_in >> 3) == h);

  // padding pair 125: clips to 0, excluded from min/max
  *(float4*)&V[125 * 128 + lane * 4] =
      make_float4(-1e30f, -1e30f, -1e30f, -1e30f);

  // 1/||x||_2 per row
  float ss = 0.0f;
  for (int j = 0; j < PAIRS; ++j) {
    const float4 a = *(const float4*)(xrow + 8 * j + 4 * h);
    ss += a.x * a.x + a.y * a.y + a.z * a.z + a.w * a.w;
  }
  ss += __shfl_xor(ss, 16, 32);
  const float rnorm = 1.0f / fmaxf(sqrtf(ss), 1e-12f);

  float lo, hi;
  sweep<true>(V, xrow, lane, h, rnorm, 0.0f, lo, hi);   // it = 0
  __syncthreads();
  float nu = bisect50(V, lane, b0, b1, b2, local, lo, hi);
  for (int it = 1; it < N_ITER; ++it) {
    __syncthreads();
    sweep<false>(V, xrow, lane, h, rnorm, nu, lo, hi);
    __syncthreads();
    nu = bisect50(V, lane, b0, b1, b2, local, lo, hi);
  }

  // losses: p_y = clip(v[row][y] - nu), loss = -log(p_y + 1e-8)
  float lsum = 0.0f;
  if (lane < 16) {
    const int yv  = y[grp * 16 + lane];
    const int j   = yv >> 3;
    const int rem = yv & 7;
    const int hh  = rem >> 2;   // which half-lane holds this column
    const int k   = rem & 3;
    const float vy = V[j * 128 + (lane + 16 * hh) * 4 + k];
    const float py = clamp01(vy - nu);
    lsum = -__logf(py + 1e-8f);
  }
#pragma unroll
  for (int d = 16; d >= 1; d >>= 1) lsum += __shfl_xor(lsum, d, 32);
  if (lane == 0) partial[grp] = lsum;
}

// Deterministic final reduction: fixed strides + fixed tree order.
__global__ __launch_bounds__(256) void reduce_partials(
    const float* __restrict__ partial, int n, float scale,
    float* __restrict__ out) {
  __shared__ float sh[256];
  const int t = threadIdx.x;
  float a = 0.0f;
  for (int i = t; i < n; i += 256) a += partial[i];
  sh[t] = a;
  __syncthreads();
  for (int off = 128; off > 0; off >>= 1) {
    if (t < off) sh[t] += sh[t + off];
    __syncthreads();
  }
  if (t == 0) out[0] = sh[0] * scale;
}

extern "C" void kernel_launch(void* const* d_in, const int* in_sizes, int n_in,
                              void* d_out, int out_size, void* d_ws,
                              size_t ws_size, hipStream_t stream) {
  const float* x = (const float*)d_in[0];
  const int*   y = (const int*)d_in[1];
  const int rows = in_sizes[1];  // 16384
  const int ngrp = rows / 16;    // 1024 single-wave workgroups
  float* partial = (float*)d_ws; // ngrp floats of scratch

  pgd_rows16<<<ngrp, 32, 0, stream>>>(x, y, partial);
  reduce_partials<<<1, 256, 0, stream>>>(partial, ngrp, 1.0f / (float)rows,
                                         (float*)d_out);
}